// Model_65730179498319
// MI455X (gfx1250) — compile-verified
//
#include <hip/hip_runtime.h>
#include <math.h>

typedef __attribute__((ext_vector_type(2))) float v2f;
typedef __attribute__((ext_vector_type(8))) float v8f;

#define B_   256
#define S_   512
#define H_   768
#define P_   32
#define H4_  3072
#define NT_  (H_ / 16)    // 48 N-tiles

// ---------------------------------------------------------------------------
// Kernel 1: fc1_w (768 x 3072 row-major) -> Wt2, the B-operand pre-swizzled
// into V_WMMA_F32_16X16X4_F32 fragment order:
//   Wt2[ ((kk*48 + nt)*32 + lane)*2 + j ] = w1[ n*3072 + 4*kk + khalf + j ]
//   with n = nt*16 + (lane&15), khalf = (lane>>4)*2, j in {0,1}
// so each GEMM k-step loads one b64 per lane, 256B contiguous per wave.
// LDS-tiled so both global reads and writes are coalesced float4s.
// Tile: 16 n-rows x 64 k-cols; grid = 48 (k-blocks) x 48 (n-tiles).
// ---------------------------------------------------------------------------
__global__ void k_swizzle_b(const float* __restrict__ w1, float* __restrict__ wt2) {
    __shared__ float tile[16][64];
    int kb = blockIdx.x / NT_;          // k-block: 64 k's
    int nt = blockIdx.x % NT_;          // n-tile: 16 n's
    int t  = threadIdx.x;               // 256 threads
    int k0 = kb * 64;
    int n0 = nt * 16;

    // read phase: thread t reads 4 consecutive k's of row (n0 + t/16)
    {
        int r  = t >> 4;
        int c4 = (t & 15) * 4;
        const float4 v = *(const float4*)(w1 + (size_t)(n0 + r) * H4_ + k0 + c4);
        tile[r][c4 + 0] = v.x;
        tile[r][c4 + 1] = v.y;
        tile[r][c4 + 2] = v.z;
        tile[r][c4 + 3] = v.w;
    }
    __syncthreads();

    // write phase: thread t writes 4 consecutive floats of one kk's fragment
    {
        int kk_l = t >> 4;              // local kk 0..15
        int s    = t & 15;
        float4 o;
        float* op = (float*)&o;
#pragma unroll
        for (int i = 0; i < 4; ++i) {
            int oidx  = s * 4 + i;      // 0..63 within this kk's 64-float chunk
            int lane  = oidx >> 1;
            int j     = oidx & 1;
            int nl    = lane & 15;
            int khalf = (lane >> 4) << 1;
            op[i] = tile[nl][4 * kk_l + khalf + j];
        }
        int kk_abs = kb * 16 + kk_l;
        *(float4*)(wt2 + (size_t)(kk_abs * NT_ + nt) * 64 + s * 4) = o;
    }
}

// ---------------------------------------------------------------------------
// Kernel 2: masked-mean pooling + CLS concat + row L2 normalize.
// One block per batch row; 256 threads, each owning 3 of the 768 H slots.
// ---------------------------------------------------------------------------
__global__ void k_build_normalize(const float* __restrict__ pvec,
                                  const float* __restrict__ mvec,
                                  const int*   __restrict__ pos,
                                  float*       __restrict__ A) {
    int b = blockIdx.x;
    int t = threadIdx.x;
    const float* pb   = pvec + (size_t)b * S_ * H_;
    const float* mb   = mvec + (size_t)b * S_ * H_;
    const int*   posb = pos + b * P_;

    float pp[3] = {0.f, 0.f, 0.f};
    float pm[3] = {0.f, 0.f, 0.f};
    int cnt = 0;
    for (int j = 0; j < P_; ++j) {
        int p = posb[j];
        if (p != -1) {
            ++cnt;
            size_t off = (size_t)p * H_;
#pragma unroll
            for (int i = 0; i < 3; ++i) {
                int h = t + i * 256;
                pp[i] += pb[off + h];
                pm[i] += mb[off + h];
            }
        }
    }
    float inv_cnt = 1.0f / (float)(cnt > 0 ? cnt : 1);

    // combined row layout: [pool_p | process_cls | pool_m | micro_cls]
    float vals[12];
#pragma unroll
    for (int i = 0; i < 3; ++i) {
        int h = t + i * 256;
        vals[i]     = pp[i] * inv_cnt;   // pool_p   @ h
        vals[3 + i] = pb[h];             // p_cls    @ 768+h
        vals[6 + i] = pm[i] * inv_cnt;   // pool_m   @ 1536+h
        vals[9 + i] = mb[h];             // m_cls    @ 2304+h
    }
    float ss = 0.f;
#pragma unroll
    for (int i = 0; i < 12; ++i) ss += vals[i] * vals[i];

    __shared__ float red[256];
    red[t] = ss;
    __syncthreads();
    for (int s = 128; s > 0; s >>= 1) {
        if (t < s) red[t] += red[t + s];
        __syncthreads();
    }
    float scale = 1.0f / fmaxf(sqrtf(red[0]), 1e-12f);

    float* Ab = A + (size_t)b * H4_;
#pragma unroll
    for (int i = 0; i < 3; ++i) {
        int h = t + i * 256;
        Ab[h]        = vals[i]     * scale;
        Ab[768  + h] = vals[3 + i] * scale;
        Ab[1536 + h] = vals[6 + i] * scale;
        Ab[2304 + h] = vals[9 + i] * scale;
    }
}

// ---------------------------------------------------------------------------
// Kernel 3: C = A(256x3072) * W^T + bias  via V_WMMA_F32_16X16X4_F32.
// One wave per 16x16 C tile; 8 waves/block; 96 blocks = 768 tiles.
// B operand comes pre-swizzled (one coalesced b64 per lane per k-step).
// 4 independent accumulators break the WMMA RAW chain (4-way ILP).
// A frag: lane<16 -> row=lane, K=khalf..khalf+1 (khalf=0); lane>=16 -> khalf=2.
// C/D: VGPR v -> M = v + (lane>=16 ? 8 : 0), N = lane%16.
// ---------------------------------------------------------------------------
__global__ void k_gemm_wmma(const float* __restrict__ A,    // 256 x 3072
                            const float* __restrict__ Bf,   // swizzled fragments
                            const float* __restrict__ bias, // 768
                            float*       __restrict__ C) {  // 256 x 768
    int wave = (blockIdx.x * blockDim.x + threadIdx.x) >> 5;
    int lane = threadIdx.x & 31;
    int mt = wave / NT_;
    int nt = wave % NT_;
    int lrow  = lane & 15;
    int khalf = (lane >> 4) << 1;   // 0 or 2

    const float* Arow = A + (size_t)(mt * 16 + lrow) * H4_ + khalf;
    const v2f*   Bfr  = (const v2f*)Bf + (size_t)nt * 32 + lane;  // + kk*48*32

    float bv = bias[nt * 16 + lrow];   // bias is per-N (per-lane)
    v8f c0, c1, c2, c3;
#pragma unroll
    for (int v = 0; v < 8; ++v) { c0[v] = bv; c1[v] = 0.f; c2[v] = 0.f; c3[v] = 0.f; }

    for (int kk = 0; kk < H4_ / 4; kk += 4) {
        __builtin_prefetch(Arow + (kk + 32) * 4, 0, 0);
        __builtin_prefetch((const float*)(Bfr + (size_t)(kk + 32) * (NT_ * 32)), 0, 0);

        v2f a0 = *(const v2f*)(Arow + (kk + 0) * 4);
        v2f a1 = *(const v2f*)(Arow + (kk + 1) * 4);
        v2f a2 = *(const v2f*)(Arow + (kk + 2) * 4);
        v2f a3 = *(const v2f*)(Arow + (kk + 3) * 4);
        v2f b0 = Bfr[(size_t)(kk + 0) * (NT_ * 32)];
        v2f b1 = Bfr[(size_t)(kk + 1) * (NT_ * 32)];
        v2f b2 = Bfr[(size_t)(kk + 2) * (NT_ * 32)];
        v2f b3 = Bfr[(size_t)(kk + 3) * (NT_ * 32)];

        c0 = __builtin_amdgcn_wmma_f32_16x16x4_f32(false, a0, false, b0, (short)0, c0, false, false);
        c1 = __builtin_amdgcn_wmma_f32_16x16x4_f32(false, a1, false, b1, (short)0, c1, false, false);
        c2 = __builtin_amdgcn_wmma_f32_16x16x4_f32(false, a2, false, b2, (short)0, c2, false, false);
        c3 = __builtin_amdgcn_wmma_f32_16x16x4_f32(false, a3, false, b3, (short)0, c3, false, false);
    }

    v8f c = (c0 + c1) + (c2 + c3);

    int nbase = nt * 16 + lrow;
    int mbase = mt * 16 + ((lane >> 4) << 3);
#pragma unroll
    for (int v = 0; v < 8; ++v)
        C[(size_t)(mbase + v) * H_ + nbase] = c[v];
}

// ---------------------------------------------------------------------------
// Kernel 4: per-feature batch-norm stats over the 256-row batch
// (C1 is 768 KB -> L2-resident; column access cost is absorbed by L2).
// ---------------------------------------------------------------------------
__global__ void k_bn_stats(const float* __restrict__ C,
                           const float* __restrict__ gamma,
                           const float* __restrict__ beta,
                           float* __restrict__ scale,
                           float* __restrict__ shift) {
    int f = blockIdx.x;
    int t = threadIdx.x;
    float x = C[(size_t)t * H_ + f];
    __shared__ float s1[256], s2[256];
    s1[t] = x;
    s2[t] = x * x;
    __syncthreads();
    for (int s = 128; s > 0; s >>= 1) {
        if (t < s) { s1[t] += s1[t + s]; s2[t] += s2[t + s]; }
        __syncthreads();
    }
    if (t == 0) {
        float mean = s1[0] * (1.0f / 256.0f);
        float var  = s2[0] * (1.0f / 256.0f) - mean * mean;
        float sc   = gamma[f] / sqrtf(var + 1e-5f);
        scale[f] = sc;
        shift[f] = beta[f] - mean * sc;
    }
}

// ---------------------------------------------------------------------------
// Kernel 5: BN apply + ReLU + final (256x768)@(768x1) projection.
// ---------------------------------------------------------------------------
__global__ void k_final(const float* __restrict__ C,
                        const float* __restrict__ scale,
                        const float* __restrict__ shift,
                        const float* __restrict__ w2,
                        const float* __restrict__ b2,
                        float* __restrict__ out) {
    int b = blockIdx.x;
    int t = threadIdx.x;
    float acc = 0.f;
#pragma unroll
    for (int i = 0; i < 3; ++i) {
        int f = t + i * 256;
        float y = C[(size_t)b * H_ + f] * scale[f] + shift[f];
        y = fmaxf(y, 0.0f);
        acc += y * w2[f];
    }
    __shared__ float red[256];
    red[t] = acc;
    __syncthreads();
    for (int s = 128; s > 0; s >>= 1) {
        if (t < s) red[t] += red[t + s];
        __syncthreads();
    }
    if (t == 0) out[b] = red[0] + b2[0];
}

// ---------------------------------------------------------------------------
extern "C" void kernel_launch(void* const* d_in, const int* in_sizes, int n_in,
                              void* d_out, int out_size, void* d_ws, size_t ws_size,
                              hipStream_t stream) {
    const float* pvec  = (const float*)d_in[0];
    const float* mvec  = (const float*)d_in[1];
    const int*   pos   = (const int*)  d_in[2];
    const float* w1    = (const float*)d_in[3];
    const float* b1    = (const float*)d_in[4];
    const float* w2    = (const float*)d_in[5];
    const float* b2    = (const float*)d_in[6];
    const float* gamma = (const float*)d_in[7];
    const float* beta  = (const float*)d_in[8];
    float* out = (float*)d_out;

    char* ws = (char*)d_ws;
    float* Wt2   = (float*)ws;  ws += (size_t)H4_ * H_ * sizeof(float);  // 9.4 MB
    float* Anorm = (float*)ws;  ws += (size_t)B_  * H4_ * sizeof(float); // 3.1 MB
    float* C1    = (float*)ws;  ws += (size_t)B_  * H_ * sizeof(float);  // 0.75 MB
    float* scl   = (float*)ws;  ws += (size_t)H_ * sizeof(float);
    float* shf   = (float*)ws;

    k_swizzle_b      <<<(H4_ / 64) * NT_,   256, 0, stream>>>(w1, Wt2);
    k_build_normalize<<<B_,                 256, 0, stream>>>(pvec, mvec, pos, Anorm);
    k_gemm_wmma      <<<(B_ / 16) * NT_ / 8, 256, 0, stream>>>(Anorm, Wt2, b1, C1);
    k_bn_stats       <<<H_,                 256, 0, stream>>>(C1, gamma, beta, scl, shf);
    k_final          <<<B_,                 256, 0, stream>>>(C1, scl, shf, w2, b2, out);
}